// Spiking_Oneway_13537736917750
// MI455X (gfx1250) — compile-verified
//
#include <hip/hip_runtime.h>
#include <hip/hip_bf16.h>

// ---- problem constants -------------------------------------------------
#define BATCH   16
#define TSTEPS  4
#define CIN     128
#define COUT    128
#define HH      56
#define WW      56
#define HWPIX   (HH*WW)            // 3136
#define PLANE_T (CIN*HWPIX)        // per (n,t) image plane stride = 401408
#define PH      58                 // padded rows
#define PWCOLS  64                 // padded cols (zero-filled 57..63)
// xcp layout: [n][pair=ci/2][ph][pw][2]  (channel pairs interleaved)
#define XCP_ELEMS  ((size_t)BATCH*(CIN/2)*PH*PWCOLS*2)    // 7,602,176 floats
#define WP_ELEMS   (8*9*32*64)                            // 147,456 floats
#define PSTR    416                // LDS per-channel-pair stride (floats)

typedef float v2f __attribute__((ext_vector_type(2)));
typedef float v8f __attribute__((ext_vector_type(8)));

// ---- prep: per-channel membrane bias  M_c = (1.875*b - mean)*inv + beta ---
__global__ void prep_m_kernel(const float* __restrict__ conv_b,
                              const float* __restrict__ gamma,
                              const float* __restrict__ beta,
                              const float* __restrict__ run_mean,
                              const float* __restrict__ run_var,
                              float* __restrict__ mc) {
    int co = threadIdx.x;
    if (co < COUT) {
        float inv = rsqrtf(run_var[co] + 1e-5f) * gamma[co];
        mc[co] = (conv_b[co] * 1.875f - run_mean[co]) * inv + beta[co];
    }
}

// ---- prep: pack weights into WMMA A-fragment layout, pre-scaled by inv/8 --
// wp[((ct*9+pos)*32 + cb)*64 + lane*2 + j] = A_co * W[co][ci][kh][kw]
//   co = ct*16 + lane%16 ; ci = cb*4 + 2*(lane/16) + j ; pos = kh*3+kw
__global__ void pack_w_kernel(const float* __restrict__ conv_w,
                              const float* __restrict__ gamma,
                              const float* __restrict__ run_var,
                              float* __restrict__ wp) {
    int idx = blockIdx.x * 256 + threadIdx.x;
    if (idx >= WP_ELEMS) return;
    int j    = idx & 1;
    int lane = (idx >> 1) & 31;
    int cb   = (idx >> 6) & 31;
    int pc   = idx >> 11;          // ct*9 + pos
    int pos  = pc % 9;
    int ct   = pc / 9;
    int co = (ct << 4) + (lane & 15);
    int ci = (cb << 2) + ((lane >> 4) << 1) + j;
    float a = rsqrtf(run_var[co] + 1e-5f) * gamma[co] * 0.125f;
    wp[idx] = conv_w[(co * CIN + ci) * 9 + pos] * a;
}

// ---- combine timesteps + zero-pad, pair-interleaved layout --------------
// xcp[(((n*64 + ci/2)*58 + ph)*64 + pw)*2 + (ci&1)]
__global__ void combine_kernel(const float* __restrict__ x,
                               float* __restrict__ xcp) {
    size_t idx = (size_t)blockIdx.x * 256 + threadIdx.x;
    if (idx >= XCP_ELEMS) return;
    int par = (int)(idx & 1);
    int pw  = (int)((idx >> 1) & 63);
    size_t t2 = idx >> 7;
    int ph = (int)(t2 % PH);
    size_t t3 = t2 / PH;
    int p  = (int)(t3 & 63);
    int n  = (int)(t3 >> 6);
    int ci = (p << 1) + par;
    float v = 0.0f;
    if (ph >= 1 && ph <= HH && pw >= 1 && pw <= WW) {
        size_t b0 = ((size_t)(n * TSTEPS) * CIN + ci) * HWPIX
                  + (size_t)(ph - 1) * WW + (pw - 1);
        v = 8.0f * x[b0]
          + 4.0f * x[b0 + (size_t)PLANE_T]
          + 2.0f * x[b0 + (size_t)2 * PLANE_T]
          +        x[b0 + (size_t)3 * PLANE_T];
    }
    xcp[idx] = v;
}

// ---- main: implicit-GEMM conv via V_WMMA_F32_16X16X4_F32 + spike scan ---
// grid = (56 rows, 16 images); block = 256 threads (8 waves).
// Wave w owns couts [16w,16w+16); 4 accumulators cover the 56-pixel row.
// LDS staged via global_load_async_to_lds_b128 (ASYNCcnt path).
__global__ __launch_bounds__(256)
void conv_spike_kernel(const float* __restrict__ xcp,
                       const float* __restrict__ wp,
                       const float* __restrict__ mc,
                       const float* __restrict__ alphap,
                       float* __restrict__ out) {
    // 32 channel-pairs x (3 rows x 64 cols x 2) with stride 416 -> 53,248 B
    __shared__ float xs[32 * PSTR];

    const int oh   = blockIdx.x;
    const int n    = blockIdx.y;
    const int tid  = threadIdx.x;
    const int wv   = tid >> 5;
    const int lane = tid & 31;
    const int laneM = lane & 15;
    const int hi    = lane >> 4;              // 0 or 1 (selects ci pair +1)

    v8f acc[4] = {};

    for (int half = 0; half < 2; ++half) {
        __syncthreads();
        // async-stage 32 pairs x 3 rows x 512B into LDS: 12 chunks per wave
        #pragma unroll
        for (int k = 0; k < 12; ++k) {
            int c = wv + (k << 3);            // 0..95 : (pair, row) chunk
            int p = c / 3;
            int r = c - p * 3;
            const float* gp = xcp
                + ((((size_t)((n << 5) + (half << 4) + p) * 2
                     + ((half & 0) ))) * 0)   /* placeholder removed below */
                ;
            (void)gp;
            const float* gsrc = xcp
                + ((((size_t)(n * 64 + (half << 5) + p)) * PH + (oh + r)) << 7)
                + (lane << 2);
            unsigned lds_off = (unsigned)(uintptr_t)
                (&xs[p * PSTR + (r << 7) + (lane << 2)]);
            unsigned long long ga = (unsigned long long)(uintptr_t)gsrc;
            asm volatile("global_load_async_to_lds_b128 %0, %1, off"
                         :: "v"(lds_off), "v"(ga) : "memory");
        }
        asm volatile("s_wait_asynccnt 0" ::: "memory");
        __syncthreads();

        for (int pos = 0; pos < 9; ++pos) {
            const int kh = pos / 3;
            const int kw = pos - kh * 3;
            const int rowoff = kh << 6;
            #pragma unroll 4
            for (int cb = 0; cb < 16; ++cb) {
                // A fragment: packed, pre-scaled weights (coalesced b64/lane)
                const v2f a = *reinterpret_cast<const v2f*>(
                    &wp[(size_t)(((wv * 9 + pos) << 5) + (half << 4) + cb) * 64
                        + (lane << 1)]);
                const int pbase = (cb * 2 + hi) * PSTR;
                #pragma unroll
                for (int t = 0; t < 4; ++t) {
                    const int col = (t << 4) + laneM + kw;
                    // single b64 LDS load: {ci, ci+1} pair already adjacent
                    const v2f b = *reinterpret_cast<const v2f*>(
                        &xs[pbase + ((rowoff + col) << 1)]);
                    acc[t] = __builtin_amdgcn_wmma_f32_16x16x4_f32(
                        false, a, false, b, (short)0, acc[t], false, false);
                }
            }
        }
    }

    // ---- epilogue: membrane bias + 4-step spike scan + store ----------
    const int cob = wv << 4;
    const int hi8 = hi << 3;
    const float thr = alphap[0] * (8.0f / 15.0f);
    const float cf0 = 0.9375f, cf1 = 0.875f, cf2 = 0.75f, cf3 = 0.5f;

    float mr[8];
    #pragma unroll
    for (int r = 0; r < 8; ++r) mr[r] = mc[cob + r + hi8];

    #pragma unroll
    for (int t = 0; t < 4; ++t) {
        const int ow = (t << 4) + laneM;
        if (ow < WW) {
            #pragma unroll
            for (int r = 0; r < 8; ++r) {
                const int co = cob + r + hi8;
                float mem = acc[t][r] + mr[r];
                size_t base = ((size_t)(n * (TSTEPS * COUT)) + co) * HWPIX
                            + (size_t)oh * WW + ow;
                float s0 = (mem >= cf0 * thr) ? thr : 0.0f; mem = (mem - s0) * 2.0f;
                float s1 = (mem >= cf1 * thr) ? thr : 0.0f; mem = (mem - s1) * 2.0f;
                float s2 = (mem >= cf2 * thr) ? thr : 0.0f; mem = (mem - s2) * 2.0f;
                float s3 = (mem >= cf3 * thr) ? thr : 0.0f;
                out[base]                              = s0;
                out[base + (size_t)COUT * HWPIX]       = s1;
                out[base + (size_t)2 * COUT * HWPIX]   = s2;
                out[base + (size_t)3 * COUT * HWPIX]   = s3;
            }
        }
    }
}

extern "C" void kernel_launch(void* const* d_in, const int* in_sizes, int n_in,
                              void* d_out, int out_size, void* d_ws, size_t ws_size,
                              hipStream_t stream) {
    (void)in_sizes; (void)n_in; (void)out_size; (void)ws_size;
    const float* x        = (const float*)d_in[0];
    const float* conv_w   = (const float*)d_in[1];
    const float* conv_b   = (const float*)d_in[2];
    const float* gamma    = (const float*)d_in[3];
    const float* beta     = (const float*)d_in[4];
    const float* run_mean = (const float*)d_in[5];
    const float* run_var  = (const float*)d_in[6];
    const float* alpha    = (const float*)d_in[7];
    float* out = (float*)d_out;

    float* ws  = (float*)d_ws;
    float* XCP = ws;                          // 7,602,176 floats (29 MB, L2-resident)
    float* WP  = XCP + XCP_ELEMS;             // 147,456 floats
    float* MC  = WP + WP_ELEMS;               // 128 floats

    prep_m_kernel<<<1, 128, 0, stream>>>(conv_b, gamma, beta, run_mean, run_var, MC);
    pack_w_kernel<<<(WP_ELEMS + 255) / 256, 256, 0, stream>>>(conv_w, gamma, run_var, WP);
    combine_kernel<<<(unsigned)((XCP_ELEMS + 255) / 256), 256, 0, stream>>>(x, XCP);

    dim3 grid(HH, BATCH);
    conv_spike_kernel<<<grid, 256, 0, stream>>>(XCP, WP, MC, alpha, out);
}